// ContrastMemory_mono_15453292331571
// MI455X (gfx1250) — compile-verified
//
#include <hip/hip_runtime.h>
#include <math.h>

typedef __attribute__((ext_vector_type(2))) float v2f;
typedef __attribute__((ext_vector_type(8))) float v8f;

#define B_ 256
#define D_ 128
#define N_ 65536
#define P_ 100
#define K_ 4096
#define KP_ 4196          // K + P
#define P2_ 10
#define OUTC_ 4106        // P2 + K
#define PSTORE_ 112       // 7 tiles * 16 (covers k < P)
#define TINV_ 14.2857142857142857f  // 1 / 0.07
#define WAVES_ 4          // waves per scoring block

// ---------------------------------------------------------------------------
// Kernel A: row L2 norms. One wave per row, float4 loads, shuffle reduction.
// Rows: [0,N) mem1 -> norm1; [N,2N) mem2 -> norm2; then v1 (B), v2 (B).
// ---------------------------------------------------------------------------
__global__ void __launch_bounds__(128)
norms_kernel(const float* __restrict__ mem1, const float* __restrict__ mem2,
             const float* __restrict__ v1,   const float* __restrict__ v2,
             float* __restrict__ norm1, float* __restrict__ norm2,
             float* __restrict__ vn1,   float* __restrict__ vn2)
{
    const int wave = threadIdx.x >> 5;
    const int lane = threadIdx.x & 31;
    const int r = blockIdx.x * WAVES_ + wave;
    const int NR = 2 * N_ + 2 * B_;
    if (r >= NR) return;

    const float* src;
    float* dst;
    if (r < N_)               { src = mem1 + (size_t)r * D_;               dst = norm1 + r; }
    else if (r < 2 * N_)      { src = mem2 + (size_t)(r - N_) * D_;        dst = norm2 + (r - N_); }
    else if (r < 2 * N_ + B_) { src = v1 + (size_t)(r - 2 * N_) * D_;      dst = vn1 + (r - 2 * N_); }
    else                      { src = v2 + (size_t)(r - 2 * N_ - B_) * D_; dst = vn2 + (r - 2 * N_ - B_); }

    const float4 x = ((const float4*)src)[lane];       // 32 lanes * 4 = 128
    float s = x.x * x.x + x.y * x.y + x.z * x.z + x.w * x.w;
    for (int m = 16; m > 0; m >>= 1) s += __shfl_xor(s, m, 32);
    if (lane == 0) *dst = sqrtf(s);
}

// ---------------------------------------------------------------------------
// Kernel B: WMMA scoring. One wave per 16-k tile; 4 waves (tiles) per block.
//   Orientation: A (16x4) rows = vectors (row0=v2[b], row1=v1[b], rest 0),
//                B (4x16) cols = 16 gathered memory rows.
//   D[m,n] = dot(vec_m, w_rowcol_n)  =>  acc[0] lanes0-15 = w·v2,
//                                        acc[1] lanes0-15 = w·v1.
//   Per-lane A source pointer chosen ONCE (branchless inner loop):
//     one ds_load_b64 (A pair) + two global_load_b32 (B pair) + v_wmma.
//   Second pass over mem2 rows reuses the same A table (acc2[1] unused).
// ---------------------------------------------------------------------------
__global__ void __launch_bounds__(WAVES_ * 32)
score_wmma_kernel(const float* __restrict__ mem1, const float* __restrict__ mem2,
                  const float* __restrict__ v1,   const float* __restrict__ v2,
                  const int* __restrict__ idx,
                  float* __restrict__ out_exp, float* __restrict__ dv1,
                  float* __restrict__ dw2)
{
    __shared__ float v1s[D_];
    __shared__ float v2s[D_];
    __shared__ float zbuf[D_ + 8];

    const int t = threadIdx.x;
    const int b = blockIdx.y;
    for (int i = t; i < D_; i += WAVES_ * 32) {
        v1s[i] = v1[b * D_ + i];
        v2s[i] = v2[b * D_ + i];
    }
    for (int i = t; i < D_ + 8; i += WAVES_ * 32) zbuf[i] = 0.f;
    __syncthreads();

    const int wave = t >> 5;
    const int lane = t & 31;
    const int tile = blockIdx.x * WAVES_ + wave;
    const int k0   = tile * 16;
    if (k0 >= KP_) return;                 // wave-uniform skip (after the only sync)

    const int col = lane & 15;             // which of the 16 k's (B column)
    const int hi  = lane >> 4;             // half: A k-pair {0,1} vs {2,3}; B rows {0,2}/{1,3}
    int k = k0 + col;
    if (k >= KP_) k = KP_ - 1;             // clamp padding cols (results discarded)
    const int row = idx[b * KP_ + k];

    // per-lane A source: row m==col of the A matrix
    const float* srcp  = (col == 0) ? v2s : (col == 1) ? v1s : zbuf;
    const float* wrow1 = mem1 + (size_t)row * D_;

    v8f acc = {0.f, 0.f, 0.f, 0.f, 0.f, 0.f, 0.f, 0.f};
#pragma unroll
    for (int c = 0; c < 32; ++c) {
        const int d = 4 * c;
        v2f a = *(const v2f*)(srcp + d + 2 * hi);           // A[m=col, d+2hi .. +1]
        v2f bb;
        bb.x = wrow1[d + hi];                               // B[row=hi,   n=col]
        bb.y = wrow1[d + 2 + hi];                           // B[row=2+hi, n=col]
        acc = __builtin_amdgcn_wmma_f32_16x16x4_f32(
            false, a, false, bb, (short)0, acc, false, false);
    }

    // D layout: VGPR j, lanes 0-15 hold D[m=j, n=lane]
    const int kk = k0 + lane;
    if (lane < 16 && kk < KP_) {
        out_exp[b * KP_ + kk] = __expf(acc[0] * TINV_);      // exp(dot(w1,v2)/T)
        if (kk < PSTORE_) dv1[b * PSTORE_ + kk] = acc[1];    // dot(w1,v1)
    }

    if (k0 < P_) {  // wave-uniform: only tiles 0..6 need s_rel numerators
        const float* wrow2 = mem2 + (size_t)row * D_;
        v8f acc2 = {0.f, 0.f, 0.f, 0.f, 0.f, 0.f, 0.f, 0.f};
#pragma unroll
        for (int c = 0; c < 32; ++c) {
            const int d = 4 * c;
            v2f a = *(const v2f*)(srcp + d + 2 * hi);
            v2f bb;
            bb.x = wrow2[d + hi];
            bb.y = wrow2[d + 2 + hi];
            acc2 = __builtin_amdgcn_wmma_f32_16x16x4_f32(
                false, a, false, bb, (short)0, acc2, false, false);
        }
        if (lane < 16 && kk < PSTORE_) dw2[b * PSTORE_ + kk] = acc2[0]; // dot(w2,v2)
    }
}

// ---------------------------------------------------------------------------
// Kernel C: per-b stable descending top-10 of diff, assemble unscaled out_cat
// into d_out, and deterministic per-b partial sum for Z.
// ---------------------------------------------------------------------------
__global__ void __launch_bounds__(128)
topk_assemble_kernel(const float* __restrict__ out_exp,
                     const float* __restrict__ dv1, const float* __restrict__ dw2,
                     const float* __restrict__ norm1, const float* __restrict__ norm2,
                     const float* __restrict__ vn1,   const float* __restrict__ vn2,
                     const int* __restrict__ idx,
                     float* __restrict__ out, float* __restrict__ bsum)
{
    const int b = blockIdx.x;
    const int t = threadIdx.x;
    __shared__ float diff[P_];
    __shared__ int   sel[P2_];
    __shared__ float red[128];

    if (t < P_) {
        int r = idx[b * KP_ + t];
        float trel = dv1[b * PSTORE_ + t] / (norm1[r] * vn1[b]);
        float srel = dw2[b * PSTORE_ + t] / (norm2[r] * vn2[b]);
        diff[t] = trel - srel;
    }
    __syncthreads();

    // stable descending rank (matches argsort(-diff): ties broken by index)
    if (t < P_) {
        float di = diff[t];
        int rk = 0;
        for (int j = 0; j < P_; ++j) {
            float dj = diff[j];
            rk += (dj > di) || (dj == di && j < t);
        }
        if (rk < P2_) sel[rk] = t;
    }
    __syncthreads();
    if (t == 0) sel[0] = 0;            // force first slot to index 0
    __syncthreads();

    float lsum = 0.f;
    if (t < P2_) {
        float v = out_exp[b * KP_ + sel[t]];
        out[(size_t)b * OUTC_ + t] = v;
        lsum += v;
    }
    for (int i = t; i < K_; i += 128) {
        float v = out_exp[b * KP_ + P_ + i];
        out[(size_t)b * OUTC_ + P2_ + i] = v;
        lsum += v;
    }
    red[t] = lsum;
    __syncthreads();
    for (int s = 64; s > 0; s >>= 1) {
        if (t < s) red[t] += red[t + s];
        __syncthreads();
    }
    if (t == 0) bsum[b] = red[0];
}

// ---------------------------------------------------------------------------
// Kernel D: reduce 256 per-b sums -> invZ = 1 / (mean * N)
// ---------------------------------------------------------------------------
__global__ void __launch_bounds__(256)
zred_kernel(const float* __restrict__ bsum, float* __restrict__ invZ)
{
    __shared__ float red[256];
    int t = threadIdx.x;
    red[t] = bsum[t];
    __syncthreads();
    for (int s = 128; s > 0; s >>= 1) {
        if (t < s) red[t] += red[t + s];
        __syncthreads();
    }
    if (t == 0) {
        float mean = red[0] / (float)((size_t)B_ * OUTC_);
        *invZ = 1.0f / (mean * (float)N_);
    }
}

// ---------------------------------------------------------------------------
// Kernel E: scale out_cat by invZ, and copy mem1 into output region 2
// ---------------------------------------------------------------------------
__global__ void __launch_bounds__(256)
finalize_kernel(float* __restrict__ out, const float* __restrict__ invZp,
                const float* __restrict__ mem1)
{
    const float invZ = *invZp;
    const size_t total1 = (size_t)B_ * OUTC_;
    const size_t total  = total1 + (size_t)N_ * D_;
    size_t i = (size_t)blockIdx.x * blockDim.x + threadIdx.x;
    const size_t stride = (size_t)gridDim.x * blockDim.x;
    for (; i < total; i += stride) {
        if (i < total1) out[i] *= invZ;
        else            out[i]  = mem1[i - total1];
    }
}

// ---------------------------------------------------------------------------
// Kernel F: overwrite rows y[b] of new_mem_v1 with normalized momentum update
// ---------------------------------------------------------------------------
__global__ void __launch_bounds__(128)
memupdate_kernel(const float* __restrict__ mem1, const float* __restrict__ v1,
                 const int* __restrict__ y, float* __restrict__ out)
{
    const int b = blockIdx.x;
    const int t = threadIdx.x;
    const int row = y[b];
    float l = mem1[(size_t)row * D_ + t] * 0.5f + v1[b * D_ + t] * 0.5f; // MOM=0.5
    __shared__ float red[128];
    red[t] = l * l;
    __syncthreads();
    for (int s = 64; s > 0; s >>= 1) {
        if (t < s) red[t] += red[t + s];
        __syncthreads();
    }
    float inv = rsqrtf(red[0]);
    out[(size_t)B_ * OUTC_ + (size_t)row * D_ + t] = l * inv;
}

// ---------------------------------------------------------------------------
extern "C" void kernel_launch(void* const* d_in, const int* in_sizes, int n_in,
                              void* d_out, int out_size, void* d_ws, size_t ws_size,
                              hipStream_t stream)
{
    // inputs: 0=epoch, 1=v1 (B,D), 2=v2 (B,D), 3=y (B), 4=idx (B,K+P),
    //         5=memory_v1 (N,D), 6=memory_v2 (N,D)
    const float* v1   = (const float*)d_in[1];
    const float* v2   = (const float*)d_in[2];
    const int*   y    = (const int*)d_in[3];
    const int*   idx  = (const int*)d_in[4];
    const float* mem1 = (const float*)d_in[5];
    const float* mem2 = (const float*)d_in[6];
    float* out = (float*)d_out;

    // workspace layout (floats)
    float* ws      = (float*)d_ws;
    float* norm1   = ws;                                 // N
    float* norm2   = norm1 + N_;                         // N
    float* vn1     = norm2 + N_;                         // B
    float* vn2     = vn1 + B_;                           // B
    float* out_exp = vn2 + B_;                           // B*KP
    float* dv1     = out_exp + (size_t)B_ * KP_;         // B*PSTORE
    float* dw2     = dv1 + (size_t)B_ * PSTORE_;         // B*PSTORE
    float* bsum    = dw2 + (size_t)B_ * PSTORE_;         // B
    float* invZ    = bsum + B_;                          // 1

    const int nrows = 2 * N_ + 2 * B_;
    norms_kernel<<<(nrows + WAVES_ - 1) / WAVES_, WAVES_ * 32, 0, stream>>>(
        mem1, mem2, v1, v2, norm1, norm2, vn1, vn2);

    const int ntiles = (KP_ + 15) / 16;                  // 263
    dim3 grid((ntiles + WAVES_ - 1) / WAVES_, B_);       // (66, 256)
    score_wmma_kernel<<<grid, WAVES_ * 32, 0, stream>>>(mem1, mem2, v1, v2, idx,
                                                        out_exp, dv1, dw2);

    topk_assemble_kernel<<<B_, 128, 0, stream>>>(out_exp, dv1, dw2,
                                                 norm1, norm2, vn1, vn2,
                                                 idx, out, bsum);

    zred_kernel<<<1, 256, 0, stream>>>(bsum, invZ);

    const size_t total = (size_t)B_ * OUTC_ + (size_t)N_ * D_;
    int blocks = (int)((total + 255) / 256);
    finalize_kernel<<<blocks, 256, 0, stream>>>(out, invZ, mem1);

    memupdate_kernel<<<B_, 128, 0, stream>>>(mem1, v1, y, out);
}